// FreLossPrecomputed_39831526703559
// MI455X (gfx1250) — compile-verified
//
#include <hip/hip_runtime.h>
#include <hip/hip_bf16.h>

// ---------------------------------------------------------------------------
// FreLoss pipeline for MI455X (gfx1250, wave32):
//   k_sph      : pred (B,N,3) -> phi, theta-pi, rho
//   k_cos      : cos DFT matrix C[n,m] = cos(2*pi*m*n/512)/512
//   k_knn_wmma : 3-NN via WMMA distance tiles. One f32 WMMA (K=4) computes a
//                full 16x16 d^2 tile: A=[phi,th,|p|^2,1], B=[-2g0,-2g1,1,|g|^2]^T.
//                Branchless top-3 selection co-executes with WMMA; half-wave
//                partials merged via __shfl_xor. -> interp (B,256,512)
//   k_gemm1    : xf = interp @ C via V_WMMA_F32_16X16X4_F32
//   k_coeff    : coeffs[b,l,m] = 2*pi * sum_k xf[b,k,m]*pct[m,l,k] -> loss terms
//   k_reduce   : deterministic single-block reduction -> loss scalar
// ---------------------------------------------------------------------------

#define NLATC 256
#define NLONC 512
#define LMAXC 50
#define MMAXC 50
#define MPADC 64
#define BATCH 2
#define NPTS  2048
#define MGRID (NLATC * NLONC)

#define PI_F     3.14159265358979323846f
#define TWOPI_F  6.28318530717958647692f

typedef __attribute__((ext_vector_type(2))) float v2f;
typedef __attribute__((ext_vector_type(8))) float v8f;

// ---------------- kernel 1: to_spherical ----------------
__global__ void k_sph(const float* __restrict__ pred,
                      float* __restrict__ phi,
                      float* __restrict__ th,
                      float* __restrict__ rho) {
  int i = blockIdx.x * blockDim.x + threadIdx.x;
  if (i >= BATCH * NPTS) return;
  float x = pred[i * 3 + 0];
  float y = pred[i * 3 + 1];
  float z = pred[i * 3 + 2];
  float r = sqrtf(x * x + y * y + z * z);
  phi[i] = atan2f(y, x);
  th[i]  = acosf(z / r) - PI_F;
  rho[i] = r;
}

// ---------------- kernel 2: cosine DFT matrix ----------------
__global__ void k_cos(float* __restrict__ cm) {
  int i = blockIdx.x * blockDim.x + threadIdx.x;
  if (i >= NLONC * MPADC) return;
  int n = i / MPADC;
  int m = i % MPADC;
  int r = (n * m) & (NLONC - 1);
  cm[i] = cosf((float)r * (TWOPI_F / (float)NLONC)) * (1.0f / (float)NLONC);
}

// ---------------- branchless sorted top-3 insert (with payload) ----------------
__device__ __forceinline__ void top3_insert(float d2, float rr,
                                            float& b0, float& b1, float& b2,
                                            float& r0, float& r1, float& r2) {
  bool lt2 = d2 < b2, lt1 = d2 < b1, lt0 = d2 < b0;
  float nb2 = lt1 ? b1 : (lt2 ? d2 : b2);
  float nr2 = lt1 ? r1 : (lt2 ? rr : r2);
  float nb1 = lt0 ? b0 : (lt1 ? d2 : b1);
  float nr1 = lt0 ? r0 : (lt1 ? rr : r1);
  b0 = lt0 ? d2 : b0;
  r0 = lt0 ? rr : r0;
  b1 = nb1; r1 = nr1;
  b2 = nb2; r2 = nr2;
}

// ---------------- kernel 3: 3-NN via WMMA distance tiles ----------------
// Block = 256 threads = 8 waves; each wave owns 16 grid points (B fragment,
// loop-invariant) and scans 2048 candidates in 128 tiles of 16 (A fragment).
// D layout: lane L & L+16 both hold grid column L&15; rows v resp. v+8.
__global__ void __launch_bounds__(256) k_knn_wmma(const float* __restrict__ phi,
                                                  const float* __restrict__ th,
                                                  const float* __restrict__ rho,
                                                  const float* __restrict__ grid,
                                                  float* __restrict__ interp) {
  __shared__ float4 spta[NPTS];  // {phi, th, |p|^2, 1.0}  (A-fragment source)
  __shared__ float  srho[NPTS];  // rho payload
  const int b = blockIdx.y;

  for (int j = threadIdx.x; j < NPTS; j += 256) {
    float ph = phi[b * NPTS + j];
    float tt = th[b * NPTS + j];
    spta[j] = make_float4(ph, tt, fmaf(ph, ph, tt * tt), 1.0f);
    srho[j] = rho[b * NPTS + j];
  }
  __syncthreads();

  const int lane = threadIdx.x & 31;
  const int wv   = threadIdx.x >> 5;
  const int l16  = lane & 15;
  const int hi   = lane >> 4;
  const int tg   = (blockIdx.x * 8 + wv) * 16 + l16;  // this lane's grid point

  // B fragment (4x16): rows K: lanes<16 -> K{0,1}, lanes>=16 -> K{2,3}
  const float g0 = grid[tg * 2 + 0];
  const float g1 = grid[tg * 2 + 1];
  v2f bb;
  bb.x = hi ? 1.0f                 : (-2.0f * g0);
  bb.y = hi ? fmaf(g0, g0, g1 * g1) : (-2.0f * g1);

  float b0 = 1e30f, b1 = 1e30f, b2 = 1e30f;
  float r0 = 0.0f,  r1 = 0.0f,  r2 = 0.0f;

  const float2* spta2 = reinterpret_cast<const float2*>(&spta[0]);
  const float4* srho4 = reinterpret_cast<const float4*>(&srho[0]);

  for (int pt = 0; pt < NPTS / 16; ++pt) {
    // A fragment: row = candidate (pt*16 + l16), cols K{0,1} or K{2,3}
    float2 av = spta2[(((pt << 4) + l16) << 1) + hi];
    v2f a;
    a.x = av.x;
    a.y = av.y;
    v8f z = {};
    v8f d = __builtin_amdgcn_wmma_f32_16x16x4_f32(
        false, a, false, bb, (short)0, z, false, false);

    const int pbase = (pt << 4) + (hi << 3);        // first candidate row
    float4 ra = srho4[pbase >> 2];                  // rho[pbase .. pbase+3]
    float4 rb = srho4[(pbase >> 2) + 1];            // rho[pbase+4 .. pbase+7]
    top3_insert(d[0], ra.x, b0, b1, b2, r0, r1, r2);
    top3_insert(d[1], ra.y, b0, b1, b2, r0, r1, r2);
    top3_insert(d[2], ra.z, b0, b1, b2, r0, r1, r2);
    top3_insert(d[3], ra.w, b0, b1, b2, r0, r1, r2);
    top3_insert(d[4], rb.x, b0, b1, b2, r0, r1, r2);
    top3_insert(d[5], rb.y, b0, b1, b2, r0, r1, r2);
    top3_insert(d[6], rb.z, b0, b1, b2, r0, r1, r2);
    top3_insert(d[7], rb.w, b0, b1, b2, r0, r1, r2);
  }

  // merge the two half-wave partial top-3s (halves hold rows 0-7 / 8-15 of each tile,
  // strict-< insert keeps this lane's own entry on ties)
  float ob0 = __shfl_xor(b0, 16, 32);
  float ob1 = __shfl_xor(b1, 16, 32);
  float ob2 = __shfl_xor(b2, 16, 32);
  float or0 = __shfl_xor(r0, 16, 32);
  float or1 = __shfl_xor(r1, 16, 32);
  float or2 = __shfl_xor(r2, 16, 32);
  top3_insert(ob0, or0, b0, b1, b2, r0, r1, r2);
  top3_insert(ob1, or1, b0, b1, b2, r0, r1, r2);
  top3_insert(ob2, or2, b0, b1, b2, r0, r1, r2);

  if (hi == 0) {
    // guard tiny negative d^2 from the |p|^2+|g|^2-2p.g expansion rounding
    float d0 = sqrtf(fmaxf(b0, 0.0f));
    float d1 = sqrtf(fmaxf(b1, 0.0f));
    float d2 = sqrtf(fmaxf(b2, 0.0f));
    float s = d0 + d1 + d2;
    interp[(size_t)b * MGRID + tg] = (r0 * d0 + r1 * d1 + r2 * d2) / s;
  }
}

// ---------------- kernel 4: xf = interp @ C via fp32 WMMA ----------------
__global__ void __launch_bounds__(32) k_gemm1(const float* __restrict__ interp,
                                              const float* __restrict__ cm,
                                              float* __restrict__ xf) {
  const int tr = blockIdx.x;
  const int tc = blockIdx.y;
  const int b  = blockIdx.z;
  const int lane = threadIdx.x;
  const int hi  = lane >> 4;
  const int l16 = lane & 15;

  const float* Arow = interp + (size_t)b * NLATC * NLONC + (size_t)(tr * 16 + l16) * NLONC;
  const float* Bcol = cm + tc * 16 + l16;

  v8f acc = {};
  for (int k0 = 0; k0 < NLONC; k0 += 4) {
    v2f a, bb;
    const float* ap = Arow + k0 + hi * 2;
    a.x = ap[0];
    a.y = ap[1];
    const float* bp = Bcol + (size_t)(k0 + hi * 2) * MPADC;
    bb.x = bp[0];
    bb.y = bp[MPADC];
    acc = __builtin_amdgcn_wmma_f32_16x16x4_f32(
        false, a, false, bb, (short)0, acc, false, false);
  }

  float* Xo = xf + (size_t)b * NLATC * MPADC + (size_t)(tr * 16) * MPADC + tc * 16;
#pragma unroll
  for (int v = 0; v < 8; ++v) {
    Xo[(size_t)(v + hi * 8) * MPADC + l16] = acc[v];
  }
}

// ---------------- kernel 5: Legendre contraction + loss terms ----------------
__global__ void k_coeff(const float* __restrict__ xf,
                        const float* __restrict__ pct,
                        const float* __restrict__ tgt,
                        const float* __restrict__ rect,
                        float* __restrict__ terms) {
  int i = blockIdx.x * blockDim.x + threadIdx.x;
  if (i >= BATCH * LMAXC * MMAXC) return;
  int m = i % MMAXC;
  int l = (i / MMAXC) % LMAXC;
  int b = i / (MMAXC * LMAXC);
  const float* xcol = xf + (size_t)b * NLATC * MPADC + m;
  const float* prow = pct + ((size_t)m * LMAXC + l) * NLATC;
  float s = 0.0f;
#pragma unroll 4
  for (int k = 0; k < NLATC; ++k) {
    s = fmaf(xcol[(size_t)k * MPADC], prow[k], s);
  }
  float c = TWOPI_F * s;
  float d = c - tgt[i];
  terms[i] = d * d * rect[l];
}

// ---------------- kernel 6: deterministic reduction -> loss ----------------
__global__ void __launch_bounds__(256) k_reduce(const float* __restrict__ terms,
                                                float* __restrict__ out) {
  __shared__ float sm[256];
  float s = 0.0f;
  for (int i = threadIdx.x; i < BATCH * LMAXC * MMAXC; i += 256) s += terms[i];
  sm[threadIdx.x] = s;
  __syncthreads();
  for (int off = 128; off > 0; off >>= 1) {
    if ((int)threadIdx.x < off) sm[threadIdx.x] += sm[threadIdx.x + off];
    __syncthreads();
  }
  if (threadIdx.x == 0) out[0] = sm[0] / (float)BATCH;
}

// ---------------- launcher ----------------
extern "C" void kernel_launch(void* const* d_in, const int* in_sizes, int n_in,
                              void* d_out, int out_size, void* d_ws, size_t ws_size,
                              hipStream_t stream) {
  const float* pred = (const float*)d_in[0];  // (2,2048,3)
  const float* tgt  = (const float*)d_in[1];  // (2,50,50)
  const float* grid = (const float*)d_in[2];  // (131072,2)
  const float* pct  = (const float*)d_in[3];  // (50,50,256)
  const float* rect = (const float*)d_in[4];  // (50,)

  float* ws     = (float*)d_ws;
  float* phi    = ws + 0;        //  4096
  float* th     = ws + 4096;     //  4096
  float* rho    = ws + 8192;     //  4096
  float* cm     = ws + 12288;    //  512*64  = 32768
  float* interp = ws + 45056;    //  2*256*512 = 262144
  float* xf     = ws + 307200;   //  2*256*64  = 32768
  float* terms  = ws + 339968;   //  5000

  k_sph<<<(BATCH * NPTS + 255) / 256, 256, 0, stream>>>(pred, phi, th, rho);
  k_cos<<<(NLONC * MPADC + 255) / 256, 256, 0, stream>>>(cm);

  dim3 gknn(MGRID / 128, BATCH);          // 8 waves/block * 16 grid pts/wave
  k_knn_wmma<<<gknn, 256, 0, stream>>>(phi, th, rho, grid, interp);

  dim3 ggemm(NLATC / 16, MPADC / 16, BATCH);
  k_gemm1<<<ggemm, 32, 0, stream>>>(interp, cm, xf);

  k_coeff<<<(BATCH * LMAXC * MMAXC + 255) / 256, 256, 0, stream>>>(xf, pct, tgt, rect, terms);
  k_reduce<<<1, 256, 0, stream>>>(terms, (float*)d_out);
}